// GCN_2886218022956
// MI455X (gfx1250) — compile-verified
//
#include <hip/hip_runtime.h>

typedef float v2f __attribute__((ext_vector_type(2)));
typedef float v8f __attribute__((ext_vector_type(8)));

#define DIN 128
#define BN_EPS 1e-5f

__device__ __forceinline__ void fatomicAdd(float* p, float v) {
  unsafeAtomicAdd(p, v);   // native global_atomic_add_f32 on gfx1250
}

// ---------------- utility ----------------
__global__ void zero_kernel(float* __restrict__ p, size_t n) {
  size_t i = (size_t)blockIdx.x * blockDim.x + threadIdx.x;
  size_t stride = (size_t)gridDim.x * blockDim.x;
  for (; i < n; i += stride) p[i] = 0.0f;
}

__global__ void degree_kernel(const int* __restrict__ dst, float* __restrict__ deg, int E) {
  int e = blockIdx.x * blockDim.x + threadIdx.x;
  if (e < E) fatomicAdd(&deg[dst[e]], 1.0f);
}

__global__ void dinv_kernel(float* __restrict__ deg, int N) {
  int i = blockIdx.x * blockDim.x + threadIdx.x;
  if (i < N) deg[i] = rsqrtf(deg[i] + 1.0f);   // +1 = self loop
}

// ---------------- WMMA GEMM ----------------
// H[N x OUTF] = act(X[N x 128]) @ W[128 x OUTF],  OUTF <= 128
// act(v) = relu(v*scale[c] + shift[c]) when scale != nullptr (fused BN+ReLU of the
// previous layer applied while staging A into LDS).
// Block = 128 threads = 4 waves. Each wave owns a 16x32 output tile (2 accumulators
// sharing one A fragment): 64 x V_WMMA_F32_16X16X4_F32 per wave over K=128.
// Both A and B tiles staged in LDS with stride 132 (+4 pad) so the 16-lane
// stride-132 fragment reads hit distinct banks and (ka,ka+1) pairs fuse to b64.
__global__ void gemm_wmma(const float* __restrict__ X, const float* __restrict__ W,
                          float* __restrict__ H, int N, int OUTF,
                          const float* __restrict__ scale, const float* __restrict__ shift) {
  __shared__ float As[16 * 132];
  __shared__ float Bs[128 * 132];        // transposed: Bs[c*132 + k]
  const int row0    = blockIdx.x * 16;
  const int colbase = blockIdx.y * 128;

  // stage A (with optional fused BN+ReLU), coalesced global reads
  for (int i = threadIdx.x; i < 16 * DIN; i += 128) {
    int r = i >> 7, c = i & 127;
    int gr = row0 + r;
    float v = 0.0f;
    if (gr < N) {
      v = X[(size_t)gr * DIN + c];
      if (scale) { v = v * scale[c] + shift[c]; v = v > 0.0f ? v : 0.0f; }
    }
    As[r * 132 + c] = v;
  }
  // stage B transposed, zero-filled beyond OUTF (guard leaves the main loop)
  for (int i = threadIdx.x; i < DIN * 128; i += 128) {
    int k = i >> 7, c = i & 127;
    int gc = colbase + c;
    float v = (gc < OUTF) ? W[(size_t)k * OUTF + gc] : 0.0f;
    Bs[c * 132 + k] = v;
  }
  __syncthreads();

  const int lane  = threadIdx.x & 31;
  const int wave  = threadIdx.x >> 5;
  const int lhalf = lane >> 4;      // 0: K pair (k,k+1)   1: K pair (k+2,k+3)
  const int lm    = lane & 15;      // A: row M;  B/D: column N
  const int c0    = wave * 32 + lm; // local column of acc0
  const int c1    = c0 + 16;        // local column of acc1

  v8f acc0 = {}, acc1 = {};
#pragma unroll
  for (int k = 0; k < DIN; k += 4) {
    const int ka = k + 2 * lhalf;
    v2f a, b0, b1;
    // 32-bit A 16x4 layout: lanes 0-15 hold K=0,1; lanes 16-31 hold K=2,3
    a.x  = As[lm * 132 + ka];
    a.y  = As[lm * 132 + ka + 1];
    // B 4x16: column per lane, mirrored K halves (read from transposed LDS tile)
    b0.x = Bs[c0 * 132 + ka];
    b0.y = Bs[c0 * 132 + ka + 1];
    b1.x = Bs[c1 * 132 + ka];
    b1.y = Bs[c1 * 132 + ka + 1];
    acc0 = __builtin_amdgcn_wmma_f32_16x16x4_f32(false, a, false, b0,
                                                 (short)0, acc0, false, false);
    acc1 = __builtin_amdgcn_wmma_f32_16x16x4_f32(false, a, false, b1,
                                                 (short)0, acc1, false, false);
  }

  // C/D layout: VGPR r -> M = r + 8*lhalf, N = lane column
  const int gc0 = colbase + c0;
  const int gc1 = colbase + c1;
#pragma unroll
  for (int r = 0; r < 8; ++r) {
    int grow = row0 + r + 8 * lhalf;
    if (grow < N) {
      if (gc0 < OUTF) H[(size_t)grow * OUTF + gc0] = acc0[r];
      if (gc1 < OUTF) H[(size_t)grow * OUTF + gc1] = acc1[r];
    }
  }
}

// ---------------- edge scatter (128-wide): wave per edge, float4 per lane ----------------
__global__ void scatter128(const float* __restrict__ H, const int* __restrict__ src,
                           const int* __restrict__ dst, const float* __restrict__ dinv,
                           float* __restrict__ agg, int E) {
  int e = blockIdx.x * (blockDim.x >> 5) + (threadIdx.x >> 5);
  if (e >= E) return;
  int ln = threadIdx.x & 31;
  int s = src[e], d = dst[e];
  float nrm = dinv[s] * dinv[d];
  const float4 v = reinterpret_cast<const float4*>(H + (size_t)s * 128)[ln];
  float* base = agg + (size_t)d * 128 + ln * 4;
  fatomicAdd(base + 0, v.x * nrm);
  fatomicAdd(base + 1, v.y * nrm);
  fatomicAdd(base + 2, v.z * nrm);
  fatomicAdd(base + 3, v.w * nrm);
}

__global__ void scatter_small(const float* __restrict__ H, const int* __restrict__ src,
                              const int* __restrict__ dst, const float* __restrict__ dinv,
                              float* __restrict__ out, int E, int F) {
  int e = blockIdx.x * (blockDim.x >> 5) + (threadIdx.x >> 5);
  if (e >= E) return;
  int ln = threadIdx.x & 31;
  int s = src[e], d = dst[e];
  float nrm = dinv[s] * dinv[d];
  for (int f = ln; f < F; f += 32)
    fatomicAdd(&out[(size_t)d * F + f], H[(size_t)s * F + f] * nrm);
}

// ---------------- fused self-loop + bias + BN statistics (128-wide) ----------------
// agg[r][f] = agg[r][f] + H[r][f]*dinv[r]^2 + bias[f]; accumulate sum/sumsq per feature.
__global__ void selfbias_stats(float* __restrict__ agg, const float* __restrict__ H,
                               const float* __restrict__ dinv, const float* __restrict__ bias,
                               float* __restrict__ stats, int N) {
  const int f    = threadIdx.x & 127;
  const int half = threadIdx.x >> 7;       // 2 threads per feature
  const int r0   = blockIdx.x * 64;
  int rend = r0 + 64; if (rend > N) rend = N;
  float s = 0.0f, q = 0.0f;
  const float bf = bias[f];
  for (int r = r0 + half; r < rend; r += 2) {
    float di = dinv[r];
    size_t idx = (size_t)r * 128 + f;
    float v = agg[idx] + H[idx] * di * di + bf;
    agg[idx] = v;
    s += v; q += v * v;
  }
  __shared__ float shs[256], shq[256];
  shs[threadIdx.x] = s; shq[threadIdx.x] = q;
  __syncthreads();
  if (threadIdx.x < 128) {
    fatomicAdd(&stats[f],       shs[threadIdx.x] + shs[threadIdx.x + 128]);
    fatomicAdd(&stats[128 + f], shq[threadIdx.x] + shq[threadIdx.x + 128]);
  }
}

// stats[0:128]=sum, [128:256]=sumsq  ->  [256:384]=scale, [384:512]=shift
__global__ void bn_finalize(float* __restrict__ stats, const float* __restrict__ g,
                            const float* __restrict__ be, float invN) {
  int f = threadIdx.x;
  float mu  = stats[f] * invN;
  float var = stats[128 + f] * invN - mu * mu;
  var = var < 0.0f ? 0.0f : var;
  float inv = rsqrtf(var + BN_EPS);
  float sc = g[f] * inv;
  stats[256 + f] = sc;
  stats[384 + f] = be[f] - mu * sc;
}

__global__ void selfbias_small(float* __restrict__ out, const float* __restrict__ H,
                               const float* __restrict__ dinv, const float* __restrict__ bias,
                               int N, int F) {
  size_t total = (size_t)N * F;
  size_t i = (size_t)blockIdx.x * blockDim.x + threadIdx.x;
  size_t stride = (size_t)gridDim.x * blockDim.x;
  for (; i < total; i += stride) {
    int r = (int)(i / F), f = (int)(i % F);
    float di = dinv[r];
    out[i] += H[i] * di * di + bias[f];
  }
}

// ---------------- launch ----------------
extern "C" void kernel_launch(void* const* d_in, const int* in_sizes, int n_in,
                              void* d_out, int out_size, void* d_ws, size_t ws_size,
                              hipStream_t stream) {
  const float* x   = (const float*)d_in[0];
  const int*   src = (const int*)d_in[1];
  const int*   dst = (const int*)d_in[2];
  const float* W1  = (const float*)d_in[3];
  const float* b1  = (const float*)d_in[4];
  const float* g1  = (const float*)d_in[5];
  const float* be1 = (const float*)d_in[6];
  const float* W2  = (const float*)d_in[7];
  const float* b2  = (const float*)d_in[8];
  const float* g2  = (const float*)d_in[9];
  const float* be2 = (const float*)d_in[10];
  const float* W3  = (const float*)d_in[11];
  const float* b3  = (const float*)d_in[12];

  const int N    = in_sizes[0] / DIN;
  const int E    = in_sizes[1];
  const int DOUT = in_sizes[12];
  float* out = (float*)d_out;

  float* ws   = (float*)d_ws;
  size_t NN   = (size_t)N * DIN;
  float* bufA = ws;                          // N*128  (H of current layer)
  float* bufB = bufA + NN;                   // N*128  (aggregation)
  float* bufC = bufB + NN;                   // N*DOUT (H3)
  float* dinv = bufC + (size_t)N * DOUT;     // N      (deg -> dinv in place)
  float* stats = dinv + N;                   // 512

  const int rowTiles = (N + 15) / 16;
  const int zgrid = 2048;

  // degree -> dinv
  zero_kernel<<<256, 256, 0, stream>>>(dinv, (size_t)N);
  degree_kernel<<<(E + 255) / 256, 256, 0, stream>>>(dst, dinv, E);
  dinv_kernel<<<(N + 255) / 256, 256, 0, stream>>>(dinv, N);

  // ---- layer 1: H1 = X @ W1 ; AGG1 = scatter + self + b1 ; BN stats ----
  gemm_wmma<<<dim3(rowTiles, 1), 128, 0, stream>>>(x, W1, bufA, N, DIN, nullptr, nullptr);
  zero_kernel<<<zgrid, 256, 0, stream>>>(bufB, NN);
  scatter128<<<(E + 7) / 8, 256, 0, stream>>>(bufA, src, dst, dinv, bufB, E);
  zero_kernel<<<1, 256, 0, stream>>>(stats, (size_t)256);
  selfbias_stats<<<(N + 63) / 64, 256, 0, stream>>>(bufB, bufA, dinv, b1, stats, N);
  bn_finalize<<<1, 128, 0, stream>>>(stats, g1, be1, 1.0f / (float)N);

  // ---- layer 2: H2 = relu(bn(AGG1)) @ W2 (BN fused into GEMM A-load) ----
  gemm_wmma<<<dim3(rowTiles, 1), 128, 0, stream>>>(bufB, W2, bufA, N, DIN, stats + 256, stats + 384);
  zero_kernel<<<zgrid, 256, 0, stream>>>(bufB, NN);
  scatter128<<<(E + 7) / 8, 256, 0, stream>>>(bufA, src, dst, dinv, bufB, E);
  zero_kernel<<<1, 256, 0, stream>>>(stats, (size_t)256);
  selfbias_stats<<<(N + 63) / 64, 256, 0, stream>>>(bufB, bufA, dinv, b2, stats, N);
  bn_finalize<<<1, 128, 0, stream>>>(stats, g2, be2, 1.0f / (float)N);

  // ---- layer 3: H3 = relu(bn(AGG2)) @ W3 ; out = scatter + self + b3 ----
  gemm_wmma<<<dim3(rowTiles, 1), 128, 0, stream>>>(bufB, W3, bufC, N, DOUT, stats + 256, stats + 384);
  zero_kernel<<<zgrid, 256, 0, stream>>>(out, (size_t)N * DOUT);
  scatter_small<<<(E + 7) / 8, 256, 0, stream>>>(bufC, src, dst, dinv, out, E, DOUT);
  selfbias_small<<<1024, 256, 0, stream>>>(out, bufC, dinv, b3, N, DOUT);
}